// DigitCapsules_29807073034743
// MI455X (gfx1250) — compile-verified
//
#include <hip/hip_runtime.h>

typedef __attribute__((ext_vector_type(2))) float v2f;
typedef __attribute__((ext_vector_type(4))) float v4f;
typedef __attribute__((ext_vector_type(8))) float v8f;

#define B_   256
#define C_   10
#define I_   1152
#define DI_  8
#define DO_  16
#define BT   16          // batch tile (WMMA N dim)
#define IC_  18          // i-chunks across grid
#define ICHUNK 64        // I_/IC_
#define WAVES 8
#define IPW  8           // i per wave = ICHUNK/WAVES
#define NPART (IC_*WAVES)   // 144 deterministic partials
#define VPITCH 164       // padded LDS pitch (multiple of 4 -> 16B aligned rows)

// ---------------------------------------------------------------------------
// Kernel 1: per (batch-tile, i-chunk) partial s[b,c,do] with on-the-fly u_hat
// via V_WMMA_F32_16X16X4_F32.  M=do, N=b, K=k(8) -> 2 chained wmma per (c,i).
// Two passes per i (logits, then scale); an asm value-barrier on the x-frags
// prevents CSE from keeping all 10 U accumulators live (register pressure).
// ---------------------------------------------------------------------------
__global__ __launch_bounds__(256) void caps_partial(
    const float* __restrict__ x,      // [B, I, 8]
    const float* __restrict__ W,      // [C, I, 1, 16, 8]
    const float* __restrict__ Vsum,   // [B, C, 16]
    float* __restrict__ sPart)        // [NPART, B, C, 16]
{
    __shared__ float VsumS[BT * VPITCH];   // Vsum tile for this batch-tile

    const int bt   = blockIdx.x / IC_;     // 0..15
    const int ic   = blockIdx.x % IC_;     // 0..17
    const int b0   = bt * BT;
    const int wave = threadIdx.x >> 5;
    const int lane = threadIdx.x & 31;
    const int bl   = lane & 15;            // N index: local batch
    const int h    = lane >> 4;            // lane half (do half / k half)

    // stage Vsum[b0..b0+15][c][do] into LDS (padded pitch to spread banks)
    for (int t = threadIdx.x; t < BT * C_ * DO_; t += 256) {
        int bb = t / (C_ * DO_);
        int j  = t - bb * (C_ * DO_);
        VsumS[bb * VPITCH + j] = Vsum[(size_t)(b0 + bb) * (C_ * DO_) + j];
    }
    __syncthreads();

    const int i0 = ic * ICHUNK + wave * IPW;
    const float* xb  = x + (size_t)(b0 + bl) * I_ * DI_;  // this lane's batch row
    const float* vsl = VsumS + bl * VPITCH + 8 * h;       // this lane's Vsum half

    v8f s[C_];
    #pragma unroll
    for (int c = 0; c < C_; ++c)
        s[c] = (v8f){0.f,0.f,0.f,0.f,0.f,0.f,0.f,0.f};

    for (int ii = 0; ii < IPW; ++ii) {
        const int i = i0 + ii;

        // B-frags (x): vgpr r + half h -> k = r + 2h (frag1: +4); contiguous pair
        v2f bx0 = *(const v2f*)(xb + i * DI_ + 2 * h);
        v2f bx1 = *(const v2f*)(xb + i * DI_ + 4 + 2 * h);

        v2f aW0[C_], aW1[C_];
        float e[C_];
        float den = 0.f;

        // ---- pass A: softmax logits for all classes (U transient) -------
        #pragma unroll
        for (int c = 0; c < C_; ++c) {
            const float* wp = W + ((size_t)(c * I_ + i) * DO_ + bl) * DI_;
            v2f a0 = *(const v2f*)(wp + 2 * h);        // A[m=do][k=2h,2h+1]
            v2f a1 = *(const v2f*)(wp + 4 + 2 * h);    // A[m=do][k=4+2h,..]
            aW0[c] = a0; aW1[c] = a1;

            v8f u = (v8f){0.f,0.f,0.f,0.f,0.f,0.f,0.f,0.f};
            u = __builtin_amdgcn_wmma_f32_16x16x4_f32(false, a0, false, bx0,
                                                      (short)0, u, false, false);
            u = __builtin_amdgcn_wmma_f32_16x16x4_f32(false, a1, false, bx1,
                                                      (short)0, u, false, false);
            // blog = <u_hat[b,c,i,:], Vsum[b,c,:]>; lane holds do = r+8h, b = bl
            v4f v0 = *(const v4f*)(vsl + c * DO_);      // 16B-aligned LDS reads
            v4f v1 = *(const v4f*)(vsl + c * DO_ + 4);
            float p = 0.f;
            p = fmaf(u[0], v0.x, p); p = fmaf(u[1], v0.y, p);
            p = fmaf(u[2], v0.z, p); p = fmaf(u[3], v0.w, p);
            p = fmaf(u[4], v1.x, p); p = fmaf(u[5], v1.y, p);
            p = fmaf(u[6], v1.z, p); p = fmaf(u[7], v1.w, p);
            p += __shfl_xor(p, 16, 32);                // add other do-half
            e[c] = __expf(p);
            den += e[c];
        }
        const float rden = 1.f / den;

        // value-barrier: make pass-B wmma chains distinct from pass-A (no CSE,
        // so the ten U accumulators are not kept live across the softmax)
        asm volatile("" : "+v"(bx0), "+v"(bx1));

        // ---- pass B: s_c += softmax_c * u_hat (recomputed) --------------
        #pragma unroll
        for (int c = 0; c < C_; ++c) {
            v8f u = (v8f){0.f,0.f,0.f,0.f,0.f,0.f,0.f,0.f};
            u = __builtin_amdgcn_wmma_f32_16x16x4_f32(false, aW0[c], false, bx0,
                                                      (short)0, u, false, false);
            u = __builtin_amdgcn_wmma_f32_16x16x4_f32(false, aW1[c], false, bx1,
                                                      (short)0, u, false, false);
            const float wgt = e[c] * rden;
            #pragma unroll
            for (int r = 0; r < 8; ++r)
                s[c][r] = fmaf(wgt, u[r], s[c][r]);
        }
    }

    // deterministic per-wave partial write: 2x b128 per class
    const int part = ic * WAVES + wave;
    float* sp = sPart + ((size_t)part * B_ + (b0 + bl)) * (C_ * DO_) + 8 * h;
    #pragma unroll
    for (int c = 0; c < C_; ++c) {
        v4f lo = {s[c][0], s[c][1], s[c][2], s[c][3]};
        v4f hi = {s[c][4], s[c][5], s[c][6], s[c][7]};
        *(v4f*)(sp + c * DO_)     = lo;
        *(v4f*)(sp + c * DO_ + 4) = hi;
    }
}

// ---------------------------------------------------------------------------
// Kernel 2: one wave per (b,c): reduce 144 partials (fixed-order lane tree),
// squash, update Vsum / write out.  Fully deterministic.
// ---------------------------------------------------------------------------
__global__ __launch_bounds__(256) void caps_squash(
    const float* __restrict__ sPart,  // [NPART, B, C, 16]
    float* __restrict__ Vsum,         // [B, C, 16]
    float* __restrict__ out,          // [B, C, 1, 16]
    int last)
{
    const int t    = (blockIdx.x * 256 + threadIdx.x) >> 5;  // (b,c) pair, 0..2559
    const int lane = threadIdx.x & 31;
    const int pg   = lane >> 2;      // partial group 0..7
    const int dq   = lane & 3;       // do-quad 0..3

    v4f acc = {0.f, 0.f, 0.f, 0.f};
    for (int p = pg; p < NPART; p += 8)                       // 18 b128 loads
        acc += *(const v4f*)(sPart + ((size_t)p * (B_ * C_) + t) * DO_ + dq * 4);

    // reduce across the 8 partial groups (lane bits 2..4), fixed order
    #pragma unroll
    for (int off = 4; off <= 16; off <<= 1) {
        acc.x += __shfl_xor(acc.x, off, 32);
        acc.y += __shfl_xor(acc.y, off, 32);
        acc.z += __shfl_xor(acc.z, off, 32);
        acc.w += __shfl_xor(acc.w, off, 32);
    }

    // |s|^2 across the 4 do-quads (lane bits 0..1)
    float sq = acc.x * acc.x + acc.y * acc.y + acc.z * acc.z + acc.w * acc.w;
    sq += __shfl_xor(sq, 1, 32);
    sq += __shfl_xor(sq, 2, 32);
    const float scale = (sq / (1.f + sq)) * rsqrtf(sq);

    if (pg == 0) {
        if (last) {
            *(v4f*)(out + (size_t)t * DO_ + dq * 4) = scale * acc;
        } else {
            v4f* d = (v4f*)(Vsum + (size_t)t * DO_ + dq * 4);
            *d += scale * acc;
        }
    }
}

__global__ __launch_bounds__(256) void caps_init(float* __restrict__ Vsum)
{
    const int t = blockIdx.x * 256 + threadIdx.x;
    if (t < B_ * C_ * DO_) Vsum[t] = 0.f;
}

// ---------------------------------------------------------------------------
extern "C" void kernel_launch(void* const* d_in, const int* in_sizes, int n_in,
                              void* d_out, int out_size, void* d_ws, size_t ws_size,
                              hipStream_t stream) {
    (void)in_sizes; (void)n_in; (void)out_size; (void)ws_size;
    const float* x = (const float*)d_in[0];   // [256,1152,8]
    const float* W = (const float*)d_in[1];   // [10,1152,1,16,8]
    float* out = (float*)d_out;               // [256,10,1,16]

    float* Vsum  = (float*)d_ws;                   // 40960 floats
    float* sPart = Vsum + (size_t)B_ * C_ * DO_;   // 144*40960 floats (~23.6 MB)

    caps_init<<<(B_ * C_ * DO_ + 255) / 256, 256, 0, stream>>>(Vsum);

    for (int it = 0; it < 3; ++it) {
        caps_partial<<<BT * IC_, 256, 0, stream>>>(x, W, Vsum, sPart);
        // 2560 (b,c) waves = 320 blocks of 8 waves
        caps_squash<<<(B_ * C_ * 32) / 256, 256, 0, stream>>>(sPart, Vsum, out, it == 2);
    }
}